// DynamiSE_644245094873
// MI455X (gfx1250) — compile-verified
//
#include <hip/hip_runtime.h>

typedef __attribute__((ext_vector_type(2))) float v2f;
typedef __attribute__((ext_vector_type(4))) float v4f;
typedef __attribute__((ext_vector_type(8))) float v8f;
typedef __attribute__((ext_vector_type(4))) int   v4i;

#define F_IN   128
#define HID    64
#define DMODEL 128
#define NSTEPS 32

// ---------------------------------------------------------------------------
// gfx1250 feature probes (compile-safe fallbacks)
// ---------------------------------------------------------------------------
#if defined(__has_builtin)
#if __has_builtin(__builtin_amdgcn_global_load_async_to_lds_b128) && \
    __has_builtin(__builtin_amdgcn_s_wait_asynccnt)
#define USE_ASYNC_LDS 1
#endif
#if __has_builtin(__builtin_amdgcn_tanhf)
#define USE_HW_TANH 1
#endif
#endif

__device__ __forceinline__ float fast_tanh(float x) {
#if defined(USE_HW_TANH)
  return __builtin_amdgcn_tanhf(x);   // V_TANH_F32 (TRANS unit)
#else
  return tanhf(x);
#endif
}

// 16B global -> LDS copy, via the CDNA5 async engine when available
__device__ __forceinline__ void copy_b128_to_lds(const float* gsrc, float* ldst) {
#if defined(USE_ASYNC_LDS)
  __builtin_amdgcn_global_load_async_to_lds_b128(
      (__attribute__((address_space(1))) v4i*)gsrc,
      (__attribute__((address_space(3))) v4i*)ldst, 0, 0);
#else
  *(v4f*)ldst = *(const v4f*)gsrc;
#endif
}
__device__ __forceinline__ void lds_copy_wait() {
#if defined(USE_ASYNC_LDS)
  __builtin_amdgcn_s_wait_asynccnt(0);   // s_wait_asynccnt 0
#endif
}

// ---------------------------------------------------------------------------
// WMMA helper: D = A(16x4) * B(4x16) + C, fp32. Layouts per CDNA5 ISA 7.12.2:
//  A: lanes 0-15 -> M=lane, VGPR0/1 = K0/K1 ; lanes 16-31 -> K2/K3
//  B: mirrored (N across lanes, K pair per lane half)
//  C/D: VGPR r -> M=r (lanes 0-15) / M=r+8 (lanes 16-31), N = lane&15
// ---------------------------------------------------------------------------
__device__ __forceinline__ v8f wmma_f32(v2f a, v2f b, v8f c) {
  return __builtin_amdgcn_wmma_f32_16x16x4_f32(false, a, false, b, (short)0, c,
                                               false, false);
}

// ============================ degree kernels ===============================
__global__ void k_deg_init(float* __restrict__ dp, float* __restrict__ dn, int n) {
  int i = blockIdx.x * blockDim.x + threadIdx.x;
  if (i < n) { dp[i] = 1.0f; dn[i] = 1.0f; }  // self-loop contributes 1
}

__global__ void k_deg_count(const int* __restrict__ dst, float* __restrict__ deg, int E) {
  int e = blockIdx.x * blockDim.x + threadIdx.x;
  if (e < E) atomicAdd(&deg[dst[e]], 1.0f);
}

__global__ void k_deg_rsqrt(float* __restrict__ dp, float* __restrict__ dn, int n) {
  int i = blockIdx.x * blockDim.x + threadIdx.x;
  if (i < n) { dp[i] = rsqrtf(dp[i]); dn[i] = rsqrtf(dn[i]); }
}

// ============================ fused encoder ================================
// Per wave: 16-row tile.  h = X*We + be (WMMA, K=128), then hw_pos = h*Wp,
// hw_neg = h*Wn (WMMA, K=64).  h round-trips through per-wave LDS scratch to
// convert C-layout -> A-layout.  W matrices read straight from global (tiny,
// L2/WGP$-resident).
__global__ void k_encoder(const float* __restrict__ X,
                          const float* __restrict__ We, const float* __restrict__ be,
                          const float* __restrict__ Wp, const float* __restrict__ Wn,
                          float* __restrict__ hwp, float* __restrict__ hwn,
                          int n_nodes) {
  extern __shared__ float sm[];
  const int lane = threadIdx.x & 31;
  const int wave = threadIdx.x >> 5;
  float* Ls = sm + wave * 2048;                       // 16x128 scratch / wave
  const int row0 = (blockIdx.x * (blockDim.x >> 5) + wave) * 16;
  if (row0 >= n_nodes) return;                        // wave-uniform branch
  const int c  = lane & 15;
  const int kh = lane >> 4;

  // stage X tile (async b128 copies, per-wave completion wait)
  #pragma unroll
  for (int t = 0; t < 16; ++t) {
    int idx = t * 32 + lane;          // 512 float4
    int r = idx >> 5, c4 = idx & 31;
    int rowc = row0 + r; if (rowc >= n_nodes) rowc = n_nodes - 1;
    copy_b128_to_lds(X + (size_t)rowc * F_IN + c4 * 4, Ls + r * F_IN + c4 * 4);
  }
  lds_copy_wait();

  // GEMM1: h(16x64) = X(16x128) * We(128x64)
  v8f h[4] = {};
  for (int kk = 0; kk < 32; ++kk) {
    v2f a = *(const v2f*)(Ls + c * F_IN + kk * 4 + kh * 2);
    int krow = kk * 4 + kh * 2;
    #pragma unroll
    for (int n = 0; n < 4; ++n) {
      v2f b = { We[krow * HID + n * 16 + c], We[(krow + 1) * HID + n * 16 + c] };
      h[n] = wmma_f32(a, b, h[n]);
    }
  }
  // h += be; restage into LDS (A-layout source for GEMM2/3)
  #pragma unroll
  for (int n = 0; n < 4; ++n) {
    float bias = be[n * 16 + c];
    #pragma unroll
    for (int r = 0; r < 8; ++r)
      Ls[(r + kh * 8) * HID + n * 16 + c] = h[n][r] + bias;
  }

  // GEMM2+3: hw_pos = h*Wp, hw_neg = h*Wn  (K=64)
  v8f p[4] = {}, q[4] = {};
  for (int kk = 0; kk < 16; ++kk) {
    v2f a = *(const v2f*)(Ls + c * HID + kk * 4 + kh * 2);
    int krow = kk * 4 + kh * 2;
    #pragma unroll
    for (int n = 0; n < 4; ++n) {
      v2f b0 = { Wp[krow * HID + n * 16 + c], Wp[(krow + 1) * HID + n * 16 + c] };
      p[n] = wmma_f32(a, b0, p[n]);
      v2f b1 = { Wn[krow * HID + n * 16 + c], Wn[(krow + 1) * HID + n * 16 + c] };
      q[n] = wmma_f32(a, b1, q[n]);
    }
  }
  #pragma unroll
  for (int n = 0; n < 4; ++n)
    #pragma unroll
    for (int r = 0; r < 8; ++r) {
      int row = row0 + r + kh * 8;
      if (row < n_nodes) {
        hwp[(size_t)row * HID + n * 16 + c] = p[n][r];
        hwn[(size_t)row * HID + n * 16 + c] = q[n][r];
      }
    }
}

// ======================= y0 = bias + self-loop term ========================
__global__ void k_y0(const float* __restrict__ hwp, const float* __restrict__ hwn,
                     const float* __restrict__ dp, const float* __restrict__ dn,
                     const float* __restrict__ bp, const float* __restrict__ bn,
                     float* __restrict__ y, int n_nodes) {
  int idx = blockIdx.x * blockDim.x + threadIdx.x;
  if (idx >= n_nodes * HID) return;
  int i = idx >> 6, f = idx & 63;
  float a = dp[i], b = dn[i];
  y[(size_t)i * DMODEL + f]        = bp[f] + hwp[idx] * a * a;
  y[(size_t)i * DMODEL + HID + f]  = bn[f] + hwn[idx] * b * b;
}

// ============================ edge scatter =================================
__global__ void k_scatter(const int* __restrict__ src, const int* __restrict__ dst,
                          const float* __restrict__ hw, const float* __restrict__ dinv,
                          float* __restrict__ y, int E, int coloff) {
  int idx = blockIdx.x * blockDim.x + threadIdx.x;
  if (idx >= E * 64) return;
  int e = idx >> 6, f = idx & 63;
  int s = src[e], d = dst[e];
  float coef = dinv[s] * dinv[d];
  atomicAdd(&y[(size_t)d * DMODEL + coloff + f], hw[(size_t)s * HID + f] * coef);
}

// ============================ ODE integrator ===============================
// One wave integrates a 16-row tile through all 32 dopri5 steps.  W_ode is
// async-staged row-major in LDS; stage inputs round-trip through per-wave LDS
// scratch (C-layout -> A-layout); all 6 RK stages live in VGPRs; tanh on the
// TRANS unit.
__device__ __forceinline__ void stage_gemm_tanh(const float* __restrict__ Wl,
                                                const float* __restrict__ Ls,
                                                const float bv[8], v8f out[8],
                                                int c, int kh) {
  v8f zero = {};
  #pragma unroll
  for (int n = 0; n < 8; ++n) out[n] = zero;
  for (int kk = 0; kk < 32; ++kk) {
    v2f a = *(const v2f*)(Ls + c * DMODEL + kk * 4 + kh * 2);
    int krow = kk * 4 + kh * 2;
    #pragma unroll
    for (int n = 0; n < 8; ++n) {
      // K-pair rows 512B apart -> ds_load_2addr_b32 candidates
      v2f b = { Wl[krow * DMODEL + n * 16 + c],
                Wl[(krow + 1) * DMODEL + n * 16 + c] };
      out[n] = wmma_f32(a, b, out[n]);
    }
  }
  #pragma unroll
  for (int n = 0; n < 8; ++n)
    #pragma unroll
    for (int r = 0; r < 8; ++r)
      out[n][r] = fast_tanh(out[n][r] + bv[n]);
}

__global__ void __launch_bounds__(128, 1)
k_ode(float* __restrict__ Y, const float* __restrict__ Wg,
      const float* __restrict__ bg, const float* __restrict__ tp, int n_nodes) {
  extern __shared__ float sm[];
  float* Wl = sm;                                   // 16384 floats, row-major
  float* bl = sm + 16384;                           // 128 floats
  float* Ls = sm + 16384 + 128 + (threadIdx.x >> 5) * 2048;  // 16x128 / wave
  const int lane = threadIdx.x & 31;
  const int c = lane & 15, kh = lane >> 4;

  // cooperative W_ode + b_ode stage via async-to-LDS engine
  for (int i = threadIdx.x * 4; i < 16384; i += blockDim.x * 4)
    copy_b128_to_lds(Wg + i, Wl + i);
  if (threadIdx.x < 32)                              // wave 0: 128 floats of b
    copy_b128_to_lds(bg + threadIdx.x * 4, bl + threadIdx.x * 4);
  lds_copy_wait();                                   // per-wave ASYNCcnt == 0
  __syncthreads();

  const float hh = (tp[1] - tp[0]) / (float)NSTEPS;
  float bv[8];
  #pragma unroll
  for (int n = 0; n < 8; ++n) bv[n] = bl[n * 16 + c];

  const int row0 = (blockIdx.x * (blockDim.x >> 5) + (threadIdx.x >> 5)) * 16;

  v8f y[8];
  #pragma unroll
  for (int n = 0; n < 8; ++n)
    #pragma unroll
    for (int r = 0; r < 8; ++r) {
      int row = row0 + r + kh * 8; if (row >= n_nodes) row = n_nodes - 1;
      y[n][r] = Y[(size_t)row * DMODEL + n * 16 + c];
    }

  v8f k0[8], k1[8], k2[8], k3[8], k4[8], k5[8];

  for (int step = 0; step < NSTEPS; ++step) {
    // ---- k0 = f(y)
    #pragma unroll
    for (int n = 0; n < 8; ++n)
      #pragma unroll
      for (int r = 0; r < 8; ++r)
        Ls[(r + kh * 8) * DMODEL + n * 16 + c] = y[n][r];
    stage_gemm_tanh(Wl, Ls, bv, k0, c, kh);
    // ---- k1 = f(y + h/5 k0)
    #pragma unroll
    for (int n = 0; n < 8; ++n)
      #pragma unroll
      for (int r = 0; r < 8; ++r)
        Ls[(r + kh * 8) * DMODEL + n * 16 + c] = y[n][r] + hh * 0.2f * k0[n][r];
    stage_gemm_tanh(Wl, Ls, bv, k1, c, kh);
    // ---- k2
    #pragma unroll
    for (int n = 0; n < 8; ++n)
      #pragma unroll
      for (int r = 0; r < 8; ++r)
        Ls[(r + kh * 8) * DMODEL + n * 16 + c] =
            y[n][r] + hh * (0.075f * k0[n][r] + 0.225f * k1[n][r]);
    stage_gemm_tanh(Wl, Ls, bv, k2, c, kh);
    // ---- k3
    #pragma unroll
    for (int n = 0; n < 8; ++n)
      #pragma unroll
      for (int r = 0; r < 8; ++r)
        Ls[(r + kh * 8) * DMODEL + n * 16 + c] =
            y[n][r] + hh * ((44.0f / 45.0f) * k0[n][r] + (-56.0f / 15.0f) * k1[n][r] +
                            (32.0f / 9.0f) * k2[n][r]);
    stage_gemm_tanh(Wl, Ls, bv, k3, c, kh);
    // ---- k4
    #pragma unroll
    for (int n = 0; n < 8; ++n)
      #pragma unroll
      for (int r = 0; r < 8; ++r)
        Ls[(r + kh * 8) * DMODEL + n * 16 + c] =
            y[n][r] + hh * ((19372.0f / 6561.0f) * k0[n][r] +
                            (-25360.0f / 2187.0f) * k1[n][r] +
                            (64448.0f / 6561.0f) * k2[n][r] +
                            (-212.0f / 729.0f) * k3[n][r]);
    stage_gemm_tanh(Wl, Ls, bv, k4, c, kh);
    // ---- k5
    #pragma unroll
    for (int n = 0; n < 8; ++n)
      #pragma unroll
      for (int r = 0; r < 8; ++r)
        Ls[(r + kh * 8) * DMODEL + n * 16 + c] =
            y[n][r] + hh * ((9017.0f / 3168.0f) * k0[n][r] +
                            (-355.0f / 33.0f) * k1[n][r] +
                            (46732.0f / 5247.0f) * k2[n][r] +
                            (49.0f / 176.0f) * k3[n][r] +
                            (-5103.0f / 18656.0f) * k4[n][r]);
    stage_gemm_tanh(Wl, Ls, bv, k5, c, kh);
    // ---- y += h * (b0 k0 + b2 k2 + b3 k3 + b4 k4 + b5 k5)   (b1 == 0)
    #pragma unroll
    for (int n = 0; n < 8; ++n)
      #pragma unroll
      for (int r = 0; r < 8; ++r)
        y[n][r] += hh * ((35.0f / 384.0f) * k0[n][r] +
                         (500.0f / 1113.0f) * k2[n][r] +
                         (125.0f / 192.0f) * k3[n][r] +
                         (-2187.0f / 6784.0f) * k4[n][r] +
                         (11.0f / 84.0f) * k5[n][r]);
  }

  #pragma unroll
  for (int n = 0; n < 8; ++n)
    #pragma unroll
    for (int r = 0; r < 8; ++r) {
      int row = row0 + r + kh * 8;
      if (row < n_nodes) Y[(size_t)row * DMODEL + n * 16 + c] = y[n][r];
    }
}

// ================================ launch ===================================
extern "C" void kernel_launch(void* const* d_in, const int* in_sizes, int n_in,
                              void* d_out, int out_size, void* d_ws, size_t ws_size,
                              hipStream_t stream) {
  const float* X  = (const float*)d_in[0];
  const int*   pe = (const int*)d_in[1];   // [2,E] pos edges (src row, dst row)
  const int*   ne = (const int*)d_in[2];   // [2,E] neg edges
  const float* tp = (const float*)d_in[3];
  const float* We = (const float*)d_in[4];
  const float* be = (const float*)d_in[5];
  const float* Wp = (const float*)d_in[6];
  const float* bp = (const float*)d_in[7];
  const float* Wn = (const float*)d_in[8];
  const float* bn = (const float*)d_in[9];
  const float* Wo = (const float*)d_in[10];
  const float* bo = (const float*)d_in[11];
  float* Y = (float*)d_out;

  const int n = in_sizes[0] / F_IN;        // 50000
  const int E = in_sizes[1] / 2;           // 800000

  // workspace layout (floats)
  float* hwp = (float*)d_ws;
  float* hwn = hwp + (size_t)n * HID;
  float* dp  = hwn + (size_t)n * HID;
  float* dn  = dp + n;

  // degrees (self-loop = init 1) -> deg^-1/2
  k_deg_init <<<(n + 255) / 256, 256, 0, stream>>>(dp, dn, n);
  k_deg_count<<<(E + 255) / 256, 256, 0, stream>>>(pe + E, dp, E);
  k_deg_count<<<(E + 255) / 256, 256, 0, stream>>>(ne + E, dn, E);
  k_deg_rsqrt<<<(n + 255) / 256, 256, 0, stream>>>(dp, dn, n);

  // encoder + hw projections (8 waves/block, 16 rows/wave)
  int gridE = (n + 128 - 1) / 128;
  k_encoder<<<gridE, 256, 8 * 2048 * sizeof(float), stream>>>(
      X, We, be, Wp, Wn, hwp, hwn, n);

  // y0 = bias + self-loop message
  k_y0<<<((n * HID) + 255) / 256, 256, 0, stream>>>(hwp, hwn, dp, dn, bp, bn, Y, n);

  // edge scatter (atomic fp32)
  int gs = (E * 64 + 255) / 256;
  k_scatter<<<gs, 256, 0, stream>>>(pe, pe + E, hwp, dp, Y, E, 0);
  k_scatter<<<gs, 256, 0, stream>>>(ne, ne + E, hwn, dn, Y, E, HID);

  // ODE: 4 waves/block, 16 rows/wave; dynamic LDS = W(64K) + b + 4x8K scratch
  int gridO = (n + 64 - 1) / 64;
  size_t shm = (size_t)(16384 + 128 + 4 * 2048) * sizeof(float);  // 98816 B
  k_ode<<<gridO, 128, shm, stream>>>(Y, Wo, bo, tp, n);
}